// SSIMLoss_19275813224970
// MI455X (gfx1250) — compile-verified
//
#include <hip/hip_runtime.h>

typedef float v2f __attribute__((ext_vector_type(2)));
typedef float v8f __attribute__((ext_vector_type(8)));
typedef int   v4i __attribute__((ext_vector_type(4)));
typedef __attribute__((address_space(1))) v4i* gv4i_p;   // global b128 ptr
typedef __attribute__((address_space(3))) v4i* lv4i_p;   // LDS b128 ptr

#define TILE    64
#define LPITCH  80                    // LDS row pitch (floats)
#define INROWS  76                    // 74 used + 2 zero rows (K padded to 28)
#define IN_ELEMS (INROWS * LPITCH)    // 6080 floats per image panel
#define VOFF    (2 * IN_ELEMS)        // 12160
#define VFIELD  (TILE * LPITCH)       // 5120 floats per vertical-conv field
#define LDS_FLOATS (VOFF + 5 * VFIELD) // 37760 floats = 151040 bytes
#define SSIM_C1 1.0e-4f
#define SSIM_C2 9.0e-4f
#define N_TOTAL 25165824.0            // 32*3*512*512

#if defined(__has_builtin)
#if __has_builtin(__builtin_amdgcn_global_load_async_to_lds_b128)
#define HAVE_ASYNC_LDS 1
#endif
#if __has_builtin(__builtin_amdgcn_s_wait_asynccnt)
#define HAVE_WAIT_ASYNC 1
#endif
#endif

// Normalized 1D Gaussian, ws=11, sigma=1.5 (matches reference g/g.sum()).
__constant__ float GW[11] = {
    1.028381e-3f, 7.598758e-3f, 3.600077e-2f, 1.093607e-1f, 2.130055e-1f,
    2.660117e-1f, 2.130055e-1f, 1.093607e-1f, 3.600077e-2f, 7.598758e-3f,
    1.028381e-3f};

__global__ void ssim_init_kernel(double* accum) { accum[0] = 0.0; }

__global__ void ssim_final_kernel(const double* accum, float* out) {
    out[0] = (float)(1.0 - accum[0] * (1.0 / N_TOTAL));
}

__global__ __launch_bounds__(256) void ssim_main_kernel(
    const float* __restrict__ img1, const float* __restrict__ img2,
    double* __restrict__ accum) {
    extern __shared__ float smem[];

    const int bid   = blockIdx.x;
    const int plane = bid >> 6;        // 96 planes
    const int tile  = bid & 63;        // 8x8 tiles of 64x64
    const int tr0   = (tile >> 3) * TILE;
    const int tc0   = (tile & 7) * TILE;
    const float* p1 = img1 + (size_t)plane * (512 * 512);
    const float* p2 = img2 + (size_t)plane * (512 * 512);

    const int tid = threadIdx.x;
    // LDS col l maps to image col (tc0 - 8 + l): left halo of 8 keeps rows
    // 16B-aligned for b128 async loads. Used cols are l = 3..76.
    // LDS row lr maps to image row (tr0 - 5 + lr); rows 0..73 used, 74/75 = 0.

    const bool interior = (tr0 >= 64) && (tr0 <= 384) && (tc0 >= 64) && (tc0 <= 384);

#ifdef HAVE_ASYNC_LDS
    if (interior) {
        // ---- Stage 0 (fast): async b128 fills, no bounds checks ----------
        int row = tid / 20;            // 20 x float4 groups per 80-float row
        int g   = tid % 20;
        const float* r1 = p1 + (size_t)(tr0 - 5) * 512 + (tc0 - 8);
        const float* r2 = p2 + (size_t)(tr0 - 5) * 512 + (tc0 - 8);
#pragma unroll
        for (int it = 0; it < 6; ++it) {
            if (row < 74) {
                const float* g1 = r1 + (size_t)row * 512 + 4 * g;
                const float* g2 = r2 + (size_t)row * 512 + 4 * g;
                float* l1 = smem + row * LPITCH + 4 * g;
                float* l2 = l1 + IN_ELEMS;
                __builtin_amdgcn_global_load_async_to_lds_b128(
                    (gv4i_p)g1, (lv4i_p)l1, 0, 0);
                __builtin_amdgcn_global_load_async_to_lds_b128(
                    (gv4i_p)g2, (lv4i_p)l2, 0, 0);
            }
            row += 12; g += 16;
            if (g >= 20) { g -= 20; row += 1; }
        }
        if (tid < 160) {               // zero the K-pad rows 74,75
            smem[74 * LPITCH + tid] = 0.f;
            smem[IN_ELEMS + 74 * LPITCH + tid] = 0.f;
        }
#ifdef HAVE_WAIT_ASYNC
        __builtin_amdgcn_s_wait_asynccnt(0);
#else
        asm volatile("s_wait_asynccnt 0" ::: "memory");
#endif
    } else
#endif
    {
        // ---- Stage 0 (edge): scalar loads with zero padding --------------
        int row = tid / LPITCH;
        int col = tid % LPITCH;
        for (int idx = tid; idx < IN_ELEMS; idx += 256) {
            int r = tr0 + row - 5;
            int c = tc0 + col - 8;
            float a = 0.f, b = 0.f;
            if (row < 74 && r >= 0 && r < 512 && c >= 0 && c < 512) {
                size_t o = (size_t)r * 512 + (size_t)c;
                a = p1[o];
                b = p2[o];
            }
            smem[idx] = a;
            smem[IN_ELEMS + idx] = b;
            row += 3; col += 16;
            if (col >= LPITCH) { col -= LPITCH; row += 1; }
        }
    }
    __syncthreads();

    const int lane = tid & 31;
    const int wv   = __builtin_amdgcn_readfirstlane(tid >> 5);  // scalar wave id
    const int m  = lane & 15;         // M for A-frags / N for B-frags
    const int hi = (lane >> 4) & 1;   // lanes 16-31 hold K+2

    // Banded 16x28 weight fragments: Band[m][k] = GW[k-m] (band 0..10).
    // Same per-lane values serve as vertical-A and horizontal-B.
    float wf0[7], wf1[7];
#pragma unroll
    for (int q = 0; q < 7; ++q) {
        int k0 = 4 * q + 2 * hi;
        int d0 = k0 - m;
        int d1 = d0 + 1;
        wf0[q] = ((unsigned)d0 <= 10u) ? GW[d0] : 0.f;
        wf1[q] = ((unsigned)d1 <= 10u) ? GW[d1] : 0.f;
    }

    // ---- Stage 1: vertical conv, all 5 fields per tile, 35 WMMAs/unit ----
    // out(16x16) = Band(16x28) x Strip(28x16); 4 row-tiles x 5 col-tiles.
    for (int u = wv; u < 20; u += 8) {
        int r0 = (u / 5) * 16;         // scalar
        int c0 = (u % 5) * 16;         // scalar; V col index j = c0 + n
        v8f a0 = {}, a1 = {}, a2 = {}, a3 = {}, a4 = {};
#pragma unroll
        for (int q = 0; q < 7; ++q) {
            int k0  = 4 * q + 2 * hi;
            int off = (r0 + k0) * LPITCH + 3 + c0 + m;  // +3: col shift (j -> lcol)
            float x1a = smem[off];
            float x2a = smem[IN_ELEMS + off];
            float x1b = smem[off + LPITCH];
            float x2b = smem[IN_ELEMS + off + LPITCH];
            v2f af = {wf0[q], wf1[q]};
            v2f b0 = {x1a, x1b};
            v2f b1 = {x2a, x2b};
            v2f b2 = {x1a * x1a, x1b * x1b};
            v2f b3 = {x2a * x2a, x2b * x2b};
            v2f b4 = {x1a * x2a, x1b * x2b};
            a0 = __builtin_amdgcn_wmma_f32_16x16x4_f32(false, af, false, b0, (short)0, a0, false, false);
            a1 = __builtin_amdgcn_wmma_f32_16x16x4_f32(false, af, false, b1, (short)0, a1, false, false);
            a2 = __builtin_amdgcn_wmma_f32_16x16x4_f32(false, af, false, b2, (short)0, a2, false, false);
            a3 = __builtin_amdgcn_wmma_f32_16x16x4_f32(false, af, false, b3, (short)0, a3, false, false);
            a4 = __builtin_amdgcn_wmma_f32_16x16x4_f32(false, af, false, b4, (short)0, a4, false, false);
        }
        float* vbase = smem + VOFF;
        int dcol = c0 + m;
#pragma unroll
        for (int v = 0; v < 8; ++v) {
            int drow = (r0 + v + 8 * hi) * LPITCH + dcol;
            vbase[0 * VFIELD + drow] = a0[v];
            vbase[1 * VFIELD + drow] = a1[v];
            vbase[2 * VFIELD + drow] = a2[v];
            vbase[3 * VFIELD + drow] = a3[v];
            vbase[4 * VFIELD + drow] = a4[v];
        }
    }
    __syncthreads();

    // ---- Stage 2: horizontal conv via WMMA + SSIM map --------------------
    float lsum = 0.f;
    for (int u = wv; u < 16; u += 8) {
        int r0 = (u >> 2) * 16;        // scalar
        int c0 = (u & 3) * 16;         // scalar
        v8f res[5];
#pragma unroll
        for (int f = 0; f < 5; ++f) {
            const float* vsrc = smem + VOFF + f * VFIELD + (r0 + m) * LPITCH + c0;
            v8f acc = {};
#pragma unroll
            for (int q = 0; q < 7; ++q) {
                int k0 = 4 * q + 2 * hi;
                v2f afrag = {vsrc[k0], vsrc[k0 + 1]};
                v2f bfrag = {wf0[q], wf1[q]};
                acc = __builtin_amdgcn_wmma_f32_16x16x4_f32(
                    false, afrag, false, bfrag, (short)0, acc, false, false);
            }
            res[f] = acc;
        }
#pragma unroll
        for (int v = 0; v < 8; ++v) {
            float mu1 = res[0][v], mu2 = res[1][v];
            float e11 = res[2][v], e22 = res[3][v], e12 = res[4][v];
            float mu1sq = mu1 * mu1;
            float mu2sq = mu2 * mu2;
            float mu12  = mu1 * mu2;
            float sg1  = e11 - mu1sq;
            float sg2  = e22 - mu2sq;
            float sg12 = e12 - mu12;
            float num = (2.f * mu12 + SSIM_C1) * (2.f * sg12 + SSIM_C2);
            float den = (mu1sq + mu2sq + SSIM_C1) * (sg1 + sg2 + SSIM_C2);
            lsum += num / den;
        }
    }

    // ---- Block reduction + global accumulate -----------------------------
    __syncthreads();
    smem[tid] = lsum;
    __syncthreads();
#pragma unroll
    for (int s = 128; s > 0; s >>= 1) {
        if (tid < s) smem[tid] += smem[tid + s];
        __syncthreads();
    }
    if (tid == 0) atomicAdd(accum, (double)smem[0]);
}

extern "C" void kernel_launch(void* const* d_in, const int* in_sizes, int n_in,
                              void* d_out, int out_size, void* d_ws,
                              size_t ws_size, hipStream_t stream) {
    const float* img1 = (const float*)d_in[0];
    const float* img2 = (const float*)d_in[1];
    float* out = (float*)d_out;
    double* accum = (double*)d_ws;

    ssim_init_kernel<<<1, 1, 0, stream>>>(accum);

    const int blocks = 96 * 64;  // 32*3 planes x (512/64)^2 tiles
    const size_t smem_bytes = LDS_FLOATS * sizeof(float);  // 151040 B
    ssim_main_kernel<<<blocks, 256, smem_bytes, stream>>>(img1, img2, accum);

    ssim_final_kernel<<<1, 1, 0, stream>>>(accum, out);
}